// FASTMultiHeadAttention_26749056320298
// MI455X (gfx1250) — compile-verified
//
#include <hip/hip_runtime.h>

// fastmax (p=2 polynomial) causal linear attention: b=4,h=8,n=4096,d=32, fp32.
// phi(x) = [1, x, vec(x x^T)]  (1057 features, padded to 1072 = 67*16)
//  pass1 : per (bh,chunk):  Mom = PhiK^T @ V  (1072x32, WMMA f32) ; Z = PhiK^T @ 1
//  prefix: exclusive prefix over chunks (causality)
//  pass2 : Num = tril(poly(QK^T)) @ V + PhiQ @ Mom ; den = rowsum(S) + PhiQ.Z
// All features computed branch-free as coef * X[j][u] * X[j][w] with X[:,32]=1,
// X[:,33]=0 so the WMMA K-loops contain no divergent control flow.
// Workspace: BH*NC*(1072*32 + 1072)*4 = 144,900,096 bytes.

typedef __attribute__((ext_vector_type(2))) float v2f;
typedef __attribute__((ext_vector_type(8))) float v8f;

namespace {
constexpr int BH  = 32;          // b*h
constexpr int SEQ = 4096;
constexpr int D   = 32;          // head dim == N of all GEMMs
constexpr int C   = 128;         // chunk length
constexpr int NC  = SEQ / C;     // 32 chunks
constexpr int DP  = 1072;        // padded feature count (67 * 16)
constexpr int MT1 = DP / 16;     // 67 feature row tiles
constexpr int KS  = 36;          // k/q LDS row stride: 32 data + [1,0,pad,pad]
constexpr int VS  = 33;          // v LDS row stride
constexpr int CHE = DP * D + DP; // floats per chunk moment record (Mom + Z)
constexpr float SCALE = 0.42044820762685725f;  // 1 / 32^0.25 (on q and k)
}

static __device__ __forceinline__ v8f wmma_f32(v2f a, v2f b, v8f c) {
  return __builtin_amdgcn_wmma_f32_16x16x4_f32(false, a, false, b, (short)0, c,
                                               false, false);
}

// Branch-free feature index map: phi[f] = coef * X[u] * X[w]  (X[32]=1, X[33]=0)
static __device__ __forceinline__ void phi_idx(int f, int& u, int& w, float& cf) {
  const int t2 = f - 33;
  u = t2 >> 5; w = t2 & 31; cf = 0.5f;              // quadratic: a2 = 0.5
  if (f < 33)    { u = f - 1; w = 32; cf = 1.0f; }  // linear:    a1 = 1
  if (f == 0)    { u = 32;    w = 32; cf = 1.0f; }  // constant:  a0 = 1
  if (f >= 1057) { u = 33;    w = 33; }             // padding -> 0
}

// ---------------------------------------------------------------------------
// Pass 1: Mom = PhiK^T @ V (1072x32), Z = PhiK^T @ ones (1072)
// ---------------------------------------------------------------------------
__global__ __launch_bounds__(256) void fastmax_p1(const float* __restrict__ Kg,
                                                  const float* __restrict__ Vg,
                                                  float* __restrict__ ws) {
  const int blk  = blockIdx.x;           // bh*NC + chunk; token offset = blk*C
  const int tid  = threadIdx.x;
  const int lane = tid & 31;
  const int wave = tid >> 5;
  const int ll   = lane & 15;
  const int lh   = lane >> 4;

  __shared__ float Ks[C][KS];
  __shared__ float Ve[C][VS];

  const float* kg = Kg + (size_t)blk * C * D;
  const float* vg = Vg + (size_t)blk * C * D;
  for (int idx = tid; idx < C * D; idx += 256) {
    const int j = idx >> 5, t = idx & 31;
    Ks[j][t] = kg[idx] * SCALE;
    Ve[j][t] = vg[idx];
  }
  for (int idx = tid; idx < C * 4; idx += 256) {   // ones/zero columns
    const int j = idx >> 2, t = idx & 3;
    Ks[j][D + t] = (t == 0) ? 1.0f : 0.0f;
  }
  __syncthreads();

  float* momc = ws + (size_t)blk * CHE;
  float* zc   = momc + DP * D;

  // GEMM: one feature row-tile per wave iteration; A generated once, 2 WMMAs.
  for (int mt = wave; mt < MT1; mt += 8) {
    int u, w; float cf;
    phi_idx(mt * 16 + ll, u, w, cf);     // per-lane loop-invariant
    v8f acc0 = {}, acc1 = {};
#pragma unroll 4
    for (int ks = 0; ks < C / 4; ++ks) {
      const int j0 = ks * 4 + 2 * lh;
      v2f a, b0, b1;
      a.x  = Ks[j0][u] * Ks[j0][w];
      a.y  = Ks[j0 + 1][u] * Ks[j0 + 1][w];
      b0.x = Ve[j0][ll];       b0.y = Ve[j0 + 1][ll];
      b1.x = Ve[j0][16 + ll];  b1.y = Ve[j0 + 1][16 + ll];
      acc0 = wmma_f32(a, b0, acc0);
      acc1 = wmma_f32(a, b1, acc1);
    }
#pragma unroll
    for (int r = 0; r < 8; ++r) {
      const size_t row = (size_t)(mt * 16 + r + 8 * lh) * D;
      momc[row + ll]      = acc0[r];
      momc[row + 16 + ll] = acc1[r];
    }
  }

  // Z[f] = sum_j phiK(f, j)  (cheap scalar reduction)
  for (int f = tid; f < DP; f += 256) {
    int u, w; float cf;
    phi_idx(f, u, w, cf);
    float z = 0.0f;
#pragma unroll 4
    for (int j = 0; j < C; ++j) z += Ks[j][u] * Ks[j][w];
    zc[f] = z;
  }
}

// ---------------------------------------------------------------------------
// Exclusive prefix over chunks (elementwise on the whole moment record).
// ---------------------------------------------------------------------------
__global__ __launch_bounds__(256) void fastmax_prefix(float* __restrict__ ws) {
  constexpr int SEG = (CHE + 15) / 16;
  const int bh = blockIdx.x;
  const int s0 = blockIdx.y * SEG;
  const int s1 = (s0 + SEG < CHE) ? (s0 + SEG) : CHE;
  float* base = ws + (size_t)bh * NC * CHE;
  for (int e = s0 + (int)threadIdx.x; e < s1; e += 256) {
    float run = 0.0f;
#pragma unroll 4
    for (int c = 0; c < NC; ++c) {
      const float t = base[(size_t)c * CHE + e];
      base[(size_t)c * CHE + e] = run;
      run += t;
    }
  }
}

// ---------------------------------------------------------------------------
// Pass 2: O = ( tril(poly(QK^T)) @ V + PhiQ @ Mom ) / ( rowsum(S) + PhiQ.Z )
// ---------------------------------------------------------------------------
__global__ __launch_bounds__(256) void fastmax_p2(const float* __restrict__ Qg,
                                                  const float* __restrict__ Kg,
                                                  const float* __restrict__ Vg,
                                                  const float* __restrict__ ws,
                                                  float* __restrict__ Og) {
  const int blk  = blockIdx.x;
  const int tid  = threadIdx.x;
  const int lane = tid & 31;
  const int mt   = tid >> 5;             // wave id == 16-row tile of the chunk
  const int ll   = lane & 15;
  const int lh   = lane >> 4;
  const int i0   = mt * 16;

  __shared__ float Qs[C][KS];
  __shared__ float Kt[C][KS];
  __shared__ float Ve[C][VS];
  __shared__ float Ss[C][C + 4];         // scores; cols 128..131 = den partials

  const float* qg = Qg + (size_t)blk * C * D;
  const float* kg = Kg + (size_t)blk * C * D;
  const float* vg = Vg + (size_t)blk * C * D;
  for (int idx = tid; idx < C * D; idx += 256) {
    const int j = idx >> 5, t = idx & 31;
    Qs[j][t] = qg[idx] * SCALE;
    Kt[j][t] = kg[idx] * SCALE;
    Ve[j][t] = vg[idx];
  }
  for (int idx = tid; idx < C * 4; idx += 256) {
    const int j = idx >> 2, t = idx & 3;
    Qs[j][D + t] = (t == 0) ? 1.0f : 0.0f;
  }
  __syncthreads();

  // -- Phase A: masked polynomial scores for this wave's 16 rows ------------
  for (int ntS = 0; ntS < 8; ++ntS) {
    v8f qk = {};
#pragma unroll
    for (int ks = 0; ks < D / 4; ++ks) {
      const int t0 = ks * 4 + 2 * lh;
      v2f a, b;
      a.x = Qs[i0 + ll][t0];        a.y = Qs[i0 + ll][t0 + 1];
      b.x = Kt[ntS * 16 + ll][t0];  b.y = Kt[ntS * 16 + ll][t0 + 1];
      qk = wmma_f32(a, b, qk);
    }
#pragma unroll
    for (int r = 0; r < 8; ++r) {
      const int i = i0 + r + 8 * lh;
      const int j = ntS * 16 + ll;
      float s = qk[r];
      s = 1.0f + s + 0.5f * s * s;
      Ss[i][j] = (j <= i) ? s : 0.0f;
    }
  }

  // -- intra denominator: row sums of S (two lanes per row) -----------------
  {
    const int row = i0 + ll;
    float part = 0.0f;
#pragma unroll 4
    for (int j = lh * 64; j < lh * 64 + 64; ++j) part += Ss[row][j];
    Ss[row][C + lh] = part;
  }

  // -- Phase B: intra numerator  Num += S @ V  (A reused across both nt) ----
  v8f acc0 = {}, acc1 = {};
#pragma unroll 4
  for (int ks = 0; ks < C / 4; ++ks) {
    const int j0 = ks * 4 + 2 * lh;
    v2f a, b0, b1;
    a.x  = Ss[i0 + ll][j0];  a.y  = Ss[i0 + ll][j0 + 1];
    b0.x = Ve[j0][ll];       b0.y = Ve[j0 + 1][ll];
    b1.x = Ve[j0][16 + ll];  b1.y = Ve[j0 + 1][16 + ll];
    acc0 = wmma_f32(a, b0, acc0);
    acc1 = wmma_f32(a, b1, acc1);
  }

  // -- Phase C: inter numerator  Num += PhiQ @ Mom(prefix) ------------------
  const float* momc = ws + (size_t)blk * CHE;
  const float* zc   = momc + DP * D;
#pragma unroll 2
  for (int kk = 0; kk < DP / 4; ++kk) {
    const int f0 = kk * 4 + 2 * lh;
    int u0, w0, u1, w1; float c0, c1;
    phi_idx(f0,     u0, w0, c0);
    phi_idx(f0 + 1, u1, w1, c1);
    v2f a, b0, b1;
    a.x  = c0 * Qs[i0 + ll][u0] * Qs[i0 + ll][w0];
    a.y  = c1 * Qs[i0 + ll][u1] * Qs[i0 + ll][w1];
    b0.x = momc[(size_t)f0 * D + ll];            b0.y = momc[(size_t)(f0 + 1) * D + ll];
    b1.x = momc[(size_t)f0 * D + 16 + ll];       b1.y = momc[(size_t)(f0 + 1) * D + 16 + ll];
    acc0 = wmma_f32(a, b0, acc0);
    acc1 = wmma_f32(a, b1, acc1);
  }

  // -- inter denominator: PhiQ . Z  (two lanes per row over feature halves) -
  {
    const int row = i0 + ll;
    float part = 0.0f;
    for (int f = lh * (DP / 2); f < (lh + 1) * (DP / 2); ++f) {
      int u, w; float cf;
      phi_idx(f, u, w, cf);
      part += cf * Qs[row][u] * Qs[row][w] * zc[f];
    }
    Ss[row][C + 2 + lh] = part;
  }

  // -- stash numerator into Ss cols 0..31, then divide ----------------------
#pragma unroll
  for (int r = 0; r < 8; ++r) {
    const int row = i0 + r + 8 * lh;
    Ss[row][ll]      = acc0[r];
    Ss[row][16 + ll] = acc1[r];
  }
  __syncthreads();

  float* og = Og + (size_t)blk * C * D;
  for (int idx = tid; idx < C * D; idx += 256) {
    const int i = idx >> 5, f = idx & 31;
    const float den = Ss[i][C] + Ss[i][C + 1] + Ss[i][C + 2] + Ss[i][C + 3];
    og[idx] = Ss[i][f] / den;
  }
}

// ---------------------------------------------------------------------------
extern "C" void kernel_launch(void* const* d_in, const int* in_sizes, int n_in,
                              void* d_out, int out_size, void* d_ws, size_t ws_size,
                              hipStream_t stream) {
  (void)in_sizes; (void)n_in; (void)out_size; (void)ws_size;
  const float* q = (const float*)d_in[0];
  const float* k = (const float*)d_in[1];
  const float* v = (const float*)d_in[2];
  // d_in[3] drop_noise unused (dropout=0); d_in[4]=mask(1), d_in[5]=p(2) baked in.
  float* o  = (float*)d_out;
  float* ws = (float*)d_ws;    // needs BH*NC*CHE*4 = 144,900,096 bytes

  fastmax_p1    <<<dim3(BH * NC), 256, 0, stream>>>(k, v, ws);
  fastmax_prefix<<<dim3(BH, 16),  256, 0, stream>>>(ws);
  fastmax_p2    <<<dim3(BH * NC), 256, 0, stream>>>(q, k, v, ws, o);
}